// FullEncoder_62534723830419
// MI455X (gfx1250) — compile-verified
//
#include <hip/hip_runtime.h>
#include <hip/hip_bf16.h>
#include <math.h>

typedef __attribute__((ext_vector_type(2))) float v2f;
typedef __attribute__((ext_vector_type(8))) float v8f;

#define BATCH 16
#define NPTS  2048
#define KNN_K 16
#define EPSF  1e-8f

// ---------------------------------------------------------------------------
// Kernel 1: features + tiny MLP encode (memory bound, one thread per node)
// features = [x, y, demand/cap, dist_to_depot, angle]
// psi' = rotate(normalize(features @ Wa^T + ba)) by theta = W2·tanh(W1·f+b1)+b2
// ---------------------------------------------------------------------------
__global__ __launch_bounds__(256) void feat_encode_kernel(
    const float* __restrict__ coords, const float* __restrict__ demands,
    const float* __restrict__ capacity,
    const float* __restrict__ Wa, const float* __restrict__ ba,
    const float* __restrict__ W1, const float* __restrict__ b1,
    const float* __restrict__ W2, const float* __restrict__ b2,
    float* __restrict__ psiOut, float* __restrict__ featOut)
{
    int gid = blockIdx.x * 256 + threadIdx.x;      // 0 .. B*N-1
    int b   = gid >> 11;                           // node gid & 2047

    const float2* cp = (const float2*)coords;
    float2 c   = cp[gid];
    float2 dep = cp[(size_t)b << 11];              // depot = node 0 of batch

    float rx = c.x - dep.x, ry = c.y - dep.y;
    float dist = sqrtf(rx * rx + ry * ry + EPSF);
    float ang  = atan2f(ry, rx);
    float dem  = demands[gid] / capacity[b];

    float f[5] = { c.x, c.y, dem, dist, ang };
    float* fo = featOut + (size_t)gid * 5;
    #pragma unroll
    for (int k = 0; k < 5; ++k) fo[k] = f[k];

    // psi = f @ Wa^T + ba, then L2-normalize (norm + EPS)
    float p0 = ba[0], p1 = ba[1];
    #pragma unroll
    for (int k = 0; k < 5; ++k) {
        p0 += f[k] * Wa[k];       // Wa row 0
        p1 += f[k] * Wa[5 + k];   // Wa row 1
    }
    float nrm = sqrtf(p0 * p0 + p1 * p1) + EPSF;
    p0 /= nrm; p1 /= nrm;

    // theta = W2 · tanh(W1·f + b1) + b2
    float theta = b2[0];
    #pragma unroll
    for (int h = 0; h < 16; ++h) {
        float acc = b1[h];
        #pragma unroll
        for (int k = 0; k < 5; ++k) acc += f[k] * W1[h * 5 + k];
        theta += tanhf(acc) * W2[h];
    }
    float cth = cosf(theta), sth = sinf(theta);
    psiOut[(size_t)gid * 2 + 0] = cth * p0 - sth * p1;
    psiOut[(size_t)gid * 2 + 1] = sth * p0 + cth * p1;
}

// ---------------------------------------------------------------------------
// Top-k helper: keep ascending sorted 16-list in registers (fully unrolled so
// arrays stay in VGPRs under wave32).
// ---------------------------------------------------------------------------
__device__ __forceinline__ void topk_insert(float (&bd)[16], int (&bi)[16],
                                            float d, int id)
{
    if (d < bd[15]) {
        bd[15] = d; bi[15] = id;
        #pragma unroll
        for (int q = 15; q > 0; --q) {
            if (bd[q] < bd[q - 1]) {
                float td = bd[q]; bd[q] = bd[q - 1]; bd[q - 1] = td;
                int   ti = bi[q]; bi[q] = bi[q - 1]; bi[q - 1] = ti;
            }
        }
    }
}

// ---------------------------------------------------------------------------
// Kernel 2: KNN via V_WMMA_F32_16X16X4_F32.
// One workgroup = one (batch, 16-row tile). 8 waves. Coords+sq staged in LDS.
// d2 tile folded into a single WMMA:  A=[x,y,sq,1], B=[-2x,-2y,1,sq]^T.
// d2 streamed through a 16x256 LDS chunk; 8 selection threads per row keep a
// register top-16, merged at the end. d2 never touches HBM.
// ---------------------------------------------------------------------------
__global__ __launch_bounds__(256) void knn_wmma_kernel(
    const float* __restrict__ coords, int* __restrict__ knnOut)
{
    __shared__ float2 sC[NPTS];          // 16 KB
    __shared__ float  sSq[NPTS];         //  8 KB
    __shared__ float  sD2[16][256];      // 16 KB  d2 chunk (16 rows x 256 cols)
    __shared__ float  sMd[16][8][16];    //  8 KB  merge distances
    __shared__ int    sMi[16][8][16];    //  8 KB  merge indices

    int bid = blockIdx.x;
    int b   = bid >> 7;                  // batch
    int rowBase = (bid & 127) << 4;      // first of 16 rows handled here

    int tid    = threadIdx.x;
    int wave   = tid >> 5;
    int lane   = tid & 31;
    int lane15 = lane & 15;
    bool hi    = lane >= 16;

    // Stage batch coords + squared norms into LDS (16 KB + 8 KB).
    const float2* cp = (const float2*)coords + ((size_t)b << 11);
    for (int i = tid; i < NPTS; i += 256) {
        float2 c = cp[i];
        sC[i]  = c;
        sSq[i] = c.x * c.x + c.y * c.y;
    }
    __syncthreads();

    // A fragment (constant for whole kernel): 32-bit A 16x4 layout —
    // lanes 0-15 hold K=0,1 ; lanes 16-31 hold K=2,3.
    int m = rowBase + lane15;
    float2 cm = sC[m];
    float sqm = sSq[m];
    v2f afrag;
    afrag.x = hi ? sqm  : cm.x;          // K0 = x  | K2 = sq
    afrag.y = hi ? 1.0f : cm.y;          // K1 = y  | K3 = 1

    // Running top-16 for selection threads (tid < 128): 8 threads per row.
    float best[16]; int bidx[16];
    #pragma unroll
    for (int q = 0; q < 16; ++q) { best[q] = 3.0e38f; bidx[q] = 0x7fffffff; }
    int selRow = tid >> 3;
    int selT   = tid & 7;

    for (int chunk = 0; chunk < 8; ++chunk) {     // 8 chunks x 256 cols
        // --- Phase A: 8 waves x 2 column tiles -> fill sD2 via WMMA ---
        #pragma unroll
        for (int sub = 0; sub < 2; ++sub) {
            int ctLocal = wave * 2 + sub;         // 0..15 within chunk
            int ct      = chunk * 16 + ctLocal;   // global column tile
            int n       = (ct << 4) + lane15;
            float2 cn = sC[n];
            float sqn = sSq[n];
            // B 4x16 layout: VGPR0 = rows K0 (lanes 0-15) / K2 (lanes 16-31),
            //                VGPR1 = rows K1 / K3.
            v2f bfrag;
            bfrag.x = hi ? 1.0f : -2.0f * cn.x;   // K0 = -2x | K2 = 1
            bfrag.y = hi ? sqn  : -2.0f * cn.y;   // K1 = -2y | K3 = sq
            v8f cacc = { 0.f, 0.f, 0.f, 0.f, 0.f, 0.f, 0.f, 0.f };
            v8f d = __builtin_amdgcn_wmma_f32_16x16x4_f32(
                false, afrag, false, bfrag, (short)0, cacc, false, false);

            // C/D layout: VGPR r -> row r (lanes 0-15) / row r+8 (lanes 16-31)
            int rOff = hi ? 8 : 0;
            int colL = (ctLocal << 4) + lane15;   // column within chunk
            int gCol = (ct << 4) + lane15;        // global column
            #pragma unroll
            for (int r = 0; r < 8; ++r) {
                float v = d[r];
                if (rowBase + r + rOff == gCol) v += 1.0e10f;  // mask diagonal
                sD2[r + rOff][colL] = v;
            }
        }
        __syncthreads();

        // --- Phase B: 128 selection threads scan the chunk ---
        if (tid < 128) {
            #pragma unroll
            for (int j = 0; j < 32; ++j) {
                int cc = selT + (j << 3);
                topk_insert(best, bidx, sD2[selRow][cc], (chunk << 8) + cc);
            }
        }
        __syncthreads();
    }

    // --- Merge 8 partial lists per row ---
    if (tid < 128) {
        #pragma unroll
        for (int q = 0; q < 16; ++q) {
            sMd[selRow][selT][q] = best[q];
            sMi[selRow][selT][q] = bidx[q];
        }
    }
    __syncthreads();

    if (tid < 16) {
        float fb[16]; int fi[16];
        #pragma unroll
        for (int q = 0; q < 16; ++q) { fb[q] = 3.0e38f; fi[q] = 0x7fffffff; }
        for (int t = 0; t < 8; ++t) {
            #pragma unroll
            for (int q = 0; q < 16; ++q)
                topk_insert(fb, fi, sMd[tid][t][q], sMi[tid][t][q]);
        }
        int* outp = knnOut + (((size_t)b << 11) + rowBase + tid) * KNN_K;
        #pragma unroll
        for (int q = 0; q < 16; ++q) outp[q] = fi[q];
    }
}

// ---------------------------------------------------------------------------
extern "C" void kernel_launch(void* const* d_in, const int* in_sizes, int n_in,
                              void* d_out, int out_size, void* d_ws, size_t ws_size,
                              hipStream_t stream)
{
    const float* coords   = (const float*)d_in[0];
    const float* demands  = (const float*)d_in[1];
    const float* capacity = (const float*)d_in[2];
    const float* Wa = (const float*)d_in[3];
    const float* ba = (const float*)d_in[4];
    const float* W1 = (const float*)d_in[5];
    const float* b1 = (const float*)d_in[6];
    const float* W2 = (const float*)d_in[7];
    const float* b2 = (const float*)d_in[8];

    float* psiOut  = (float*)d_out;                      // 16*2048*2
    float* featOut = psiOut + BATCH * NPTS * 2;          // 16*2048*5
    int*   knnOut  = (int*)(featOut + BATCH * NPTS * 5); // 16*2048*16 (i32 bits)

    feat_encode_kernel<<<(BATCH * NPTS) / 256, 256, 0, stream>>>(
        coords, demands, capacity, Wa, ba, W1, b1, W2, b2, psiOut, featOut);

    knn_wmma_kernel<<<BATCH * (NPTS / 16), 256, 0, stream>>>(coords, knnOut);
}